// HO_GCN_89635967467586
// MI455X (gfx1250) — compile-verified
//
#include <hip/hip_runtime.h>
#include <math.h>

// ---------------------------------------------------------------------------
// GCN pipeline for MI455X (gfx1250, wave32).
// FO branch of the reference is dead code (output depends only on the HO
// branch + bipartite operator + MLP) and is eliminated.
// Dense transforms: V_WMMA_F32_16X16X4_F32 (fp32-exact).
// Edge aggregation: B128 gathers + native global_atomic_add_f32 into
// L2-resident accumulators (all node arrays fit in the 192MB L2).
// ---------------------------------------------------------------------------

typedef float v2f __attribute__((ext_vector_type(2)));
typedef float v8f __attribute__((ext_vector_type(8)));

#if defined(__has_builtin)
#if __has_builtin(__builtin_amdgcn_wmma_f32_16x16x4_f32)
#define HAVE_WMMA_F32 1
#endif
#endif

__device__ __forceinline__ float elu1(float x) {
    return x > 0.0f ? x : expm1f(x);
}

__device__ __forceinline__ void atomAddF(float* p, float v) {
    __hip_atomic_fetch_add(p, v, __ATOMIC_RELAXED, __HIP_MEMORY_SCOPE_AGENT);
}

// ---------------- degree + per-edge coefficient -----------------------------

__global__ void fill_ones_kernel(float* __restrict__ p, int n) {
    int i = blockIdx.x * blockDim.x + threadIdx.x;
    if (i < n) p[i] = 1.0f;
}

__global__ void deg_scatter_kernel(const int* __restrict__ dst,
                                   const float* __restrict__ w,
                                   float* __restrict__ deg, int E) {
    int e = blockIdx.x * blockDim.x + threadIdx.x;
    if (e < E) atomAddF(&deg[dst[e]], w[e]);
}

// coef[e] = rsqrt(deg[src]) * w * rsqrt(deg[dst])   (shared by both layers)
__global__ void edge_coef_kernel(const int* __restrict__ src,
                                 const int* __restrict__ dst,
                                 const float* __restrict__ w,
                                 const float* __restrict__ deg,
                                 float* __restrict__ coef, int E) {
    int e = blockIdx.x * blockDim.x + threadIdx.x;
    if (e < E) coef[e] = rsqrtf(deg[src[e]]) * w[e] * rsqrtf(deg[dst[e]]);
}

// ---------------- WMMA GEMM: C[M,N] = A[M,K] @ W[K,N] (+ bias) --------------
// One wave32 computes one 16x16 tile. M,N multiples of 16; K multiple of 4.

__global__ void gemm_wmma_f32_kernel(const float* __restrict__ A,
                                     const float* __restrict__ W,
                                     const float* __restrict__ bias,
                                     float* __restrict__ C,
                                     int M, int K, int N) {
    const int wave = blockIdx.x * (blockDim.x >> 5) + (threadIdx.x >> 5);
    const int lane = threadIdx.x & 31;
    const int ntc  = N >> 4;                 // tiles along N
    const int tiles = (M >> 4) * ntc;
    if (wave >= tiles) return;               // wave-uniform: EXEC stays all-1s
    const int rt = wave / ntc;
    const int ct = wave - rt * ntc;
    const int r  = lane & 15;
    const int hi = lane >> 4;
    const int col = (ct << 4) + r;

    v8f acc = {};
#if defined(HAVE_WMMA_F32)
    // A frag: v0 = A[row][ko + 2*hi], v1 = A[row][ko + 2*hi + 1]
    // B frag: v0 = W[ko + 2*hi][col], v1 = W[ko + 2*hi + 1][col]
    const int row = (rt << 4) + r;
    for (int ko = 0; ko < K; ko += 4) {
        const int ka = ko + (hi << 1);
        // row*K + ka is even for all our shapes -> 8B-aligned 64-bit load
        v2f a = *(const v2f*)(A + row * K + ka);
        v2f b;
        b.x = W[ka * N + col];
        b.y = W[(ka + 1) * N + col];
        acc = __builtin_amdgcn_wmma_f32_16x16x4_f32(
            false, a, false, b, (short)0, acc, false, false);
    }
#else
    for (int i = 0; i < 8; ++i) {
        const int rr = (rt << 4) + i + (hi << 3);
        float s = 0.0f;
        for (int k = 0; k < K; ++k) s = fmaf(A[rr * K + k], W[k * N + col], s);
        acc[i] = s;
    }
#endif
    const float bv = bias ? bias[col] : 0.0f;
    // D layout: VGPR i holds M = rt*16 + i + 8*hi, N = col
    #pragma unroll
    for (int i = 0; i < 8; ++i) {
        const int rr = (rt << 4) + i + (hi << 3);
        C[rr * N + col] = acc[i] + bv;
    }
}

// ---------------- edge scatter: agg[dst] += xw[src] * coef ------------------
// Thread handles 4 consecutive features: one B128 gather + 4 f32 atomics.

template <int LOGF>
__global__ void gcn_scatter_kernel(const int* __restrict__ src,
                                   const int* __restrict__ dst,
                                   const float* __restrict__ coef,
                                   const float4* __restrict__ xw4,
                                   float* __restrict__ agg,
                                   int E) {
    constexpr int LOGQ = LOGF - 2;           // quads per node row
    const int idx = blockIdx.x * blockDim.x + threadIdx.x;
    if (idx >= (E << LOGQ)) return;
    const int e = idx >> LOGQ;
    const int q = idx & ((1 << LOGQ) - 1);
    const int s = src[e];
    const int d = dst[e];
    const float cf = coef[e];
    const float4 x = xw4[((size_t)s << LOGQ) + q];
    float* ap = agg + ((size_t)d << LOGF) + (q << 2);
    atomAddF(ap + 0, x.x * cf);
    atomAddF(ap + 1, x.y * cf);
    atomAddF(ap + 2, x.z * cf);
    atomAddF(ap + 3, x.w * cf);
}

// ---------------- node finish: h = elu(agg + xw/deg + b) --------------------

template <int LOGF>
__global__ void node_finish_kernel(float4* h,            // may alias agg
                                   const float4* agg,
                                   const float4* __restrict__ xw,
                                   const float* __restrict__ deg,
                                   const float* __restrict__ bias,
                                   int n) {
    constexpr int LOGQ = LOGF - 2;
    const int idx = blockIdx.x * blockDim.x + threadIdx.x;
    if (idx >= (n << LOGQ)) return;
    const int i = idx >> LOGQ;
    const int q = idx & ((1 << LOGQ) - 1);
    const float di = 1.0f / deg[i];
    const float4 a = agg[idx];
    const float4 x = xw[idx];
    float4 r;
    r.x = elu1(a.x + x.x * di + bias[(q << 2) + 0]);
    r.y = elu1(a.y + x.y * di + bias[(q << 2) + 1]);
    r.z = elu1(a.z + x.z * di + bias[(q << 2) + 2]);
    r.w = elu1(a.w + x.w * di + bias[(q << 2) + 3]);
    h[idx] = r;
}

// ---------------- bipartite scatter: agg[dst] += m[src] (F = 32) ------------

__global__ void bi_scatter_kernel(const int* __restrict__ src,
                                  const int* __restrict__ dst,
                                  const float4* __restrict__ m4,
                                  float* __restrict__ agg,
                                  int E) {
    const int idx = blockIdx.x * blockDim.x + threadIdx.x;
    if (idx >= (E << 3)) return;             // 8 quads per 32-wide row
    const int e = idx >> 3;
    const int q = idx & 7;
    const float4 x = m4[((size_t)src[e] << 3) + q];
    float* ap = agg + ((size_t)dst[e] << 5) + (q << 2);
    atomAddF(ap + 0, x.x);
    atomAddF(ap + 1, x.y);
    atomAddF(ap + 2, x.z);
    atomAddF(ap + 3, x.w);
}

// ---------------- output MLP: out = elu(agg) @ mlp_w + mlp_b ----------------

__global__ void out_mlp_kernel(const float* __restrict__ agg,
                               const float* __restrict__ w,   // [32,10]
                               const float* __restrict__ b,   // [10]
                               float* __restrict__ out,
                               int nfo) {
    const int i = blockIdx.x * blockDim.x + threadIdx.x;
    if (i >= nfo) return;
    float e[32];
    #pragma unroll
    for (int c = 0; c < 32; ++c) e[c] = elu1(agg[i * 32 + c]);
    #pragma unroll
    for (int j = 0; j < 10; ++j) {
        float s = b[j];
        #pragma unroll
        for (int c = 0; c < 32; ++c) s = fmaf(e[c], w[c * 10 + j], s);
        out[i * 10 + j] = s;
    }
}

// ---------------------------------------------------------------------------

static inline int cdiv(long long a, long long b) { return (int)((a + b - 1) / b); }

extern "C" void kernel_launch(void* const* d_in, const int* in_sizes, int n_in,
                              void* d_out, int out_size, void* d_ws, size_t ws_size,
                              hipStream_t stream) {
    const float* x_ho   = (const float*)d_in[0];
    const int*   ho_src = (const int*)  d_in[2];
    const int*   ho_dst = (const int*)  d_in[3];
    const float* ho_w   = (const float*)d_in[4];
    const int*   bi_src = (const int*)  d_in[8];
    const int*   bi_dst = (const int*)  d_in[9];
    const float* W220   = (const float*)d_in[12];
    const float* b220   = (const float*)d_in[13];
    const float* W221   = (const float*)d_in[14];
    const float* b221   = (const float*)d_in[15];
    const float* lin1_w = (const float*)d_in[20];
    const float* lin1_b = (const float*)d_in[21];
    const float* mlp_w  = (const float*)d_in[24];
    const float* mlp_b  = (const float*)d_in[25];

    const int NHO = in_sizes[0] / 60;   // 400000
    const int NFO = in_sizes[1] / 32;   // 100000
    const int EHO = in_sizes[2];        // 4000000
    const int EBI = in_sizes[8];        // 800000

    // workspace partition (floats)
    float* bufA  = (float*)d_ws;                 // NHO*32 : xw / m
    float* bufB  = bufA + (size_t)NHO * 32;      // NHO*32 : agg / h
    float* aggbi = bufB + (size_t)NHO * 32;      // NFO*32
    float* deg   = aggbi + (size_t)NFO * 32;     // NHO
    float* coef  = deg + (size_t)NHO;            // EHO
    const size_t need =
        ((size_t)NHO * 65 + (size_t)NFO * 32 + (size_t)EHO) * sizeof(float);
    if (ws_size < need) return;

    const int T = 256;

    // 1) degrees + per-edge normalization coefficient
    fill_ones_kernel<<<cdiv(NHO, T), T, 0, stream>>>(deg, NHO);
    deg_scatter_kernel<<<cdiv(EHO, T), T, 0, stream>>>(ho_dst, ho_w, deg, EHO);
    edge_coef_kernel<<<cdiv(EHO, T), T, 0, stream>>>(ho_src, ho_dst, ho_w, deg,
                                                     coef, EHO);

    // 2) layer 1: xw = x_ho @ W220  (400000x60x16)
    {
        int tiles = (NHO / 16) * (16 / 16);
        gemm_wmma_f32_kernel<<<cdiv(tiles, 4), 128, 0, stream>>>(
            x_ho, W220, nullptr, bufA, NHO, 60, 16);
    }
    hipMemsetAsync(bufB, 0, (size_t)NHO * 16 * sizeof(float), stream);
    gcn_scatter_kernel<4><<<cdiv((long long)EHO << 2, T), T, 0, stream>>>(
        ho_src, ho_dst, coef, (const float4*)bufA, bufB, EHO);
    node_finish_kernel<4><<<cdiv((long long)NHO << 2, T), T, 0, stream>>>(
        (float4*)bufB, (const float4*)bufB, (const float4*)bufA, deg, b220, NHO);

    // 3) layer 2: xw = h @ W221  (400000x16x32)
    {
        int tiles = (NHO / 16) * (32 / 16);
        gemm_wmma_f32_kernel<<<cdiv(tiles, 4), 128, 0, stream>>>(
            bufB, W221, nullptr, bufA, NHO, 16, 32);
    }
    hipMemsetAsync(bufB, 0, (size_t)NHO * 32 * sizeof(float), stream);
    gcn_scatter_kernel<5><<<cdiv((long long)EHO << 3, T), T, 0, stream>>>(
        ho_src, ho_dst, coef, (const float4*)bufA, bufB, EHO);
    node_finish_kernel<5><<<cdiv((long long)NHO << 3, T), T, 0, stream>>>(
        (float4*)bufB, (const float4*)bufB, (const float4*)bufA, deg, b221, NHO);

    // 4) bipartite: m = h @ lin1_w + lin1_b ; agg_bi = segment_sum(m[bi_src])
    {
        int tiles = (NHO / 16) * (32 / 16);
        gemm_wmma_f32_kernel<<<cdiv(tiles, 4), 128, 0, stream>>>(
            bufB, lin1_w, lin1_b, bufA, NHO, 32, 32);
    }
    hipMemsetAsync(aggbi, 0, (size_t)NFO * 32 * sizeof(float), stream);
    bi_scatter_kernel<<<cdiv((long long)EBI << 3, T), T, 0, stream>>>(
        bi_src, bi_dst, (const float4*)bufA, aggbi, EBI);

    // 5) out = elu(agg_bi) @ mlp_w + mlp_b
    out_mlp_kernel<<<cdiv(NFO, 128), 128, 0, stream>>>(
        aggbi, mlp_w, mlp_b, (float*)d_out, NFO);
}